// Model_32873679684312
// MI455X (gfx1250) — compile-verified
//
#include <hip/hip_runtime.h>
#include <hip/hip_bf16.h>

// Test-time-training scan: B=2048 independent rows, each carries its own MLP
// (2 -> 64 -> 64 -> 1, tanh) through 783 sequential steps of 4 inner SGD steps.
// One wave32 per row. W2 (64x64) lives in LDS (stride-66 padded: conflict-free
// for row access (b64), column access, and WMMA tile staging). Inner SGD uses
// the exact dual form so W2 is frozen during the 4 steps; the deferred update
// is a rank-4 GEMM done with V_WMMA_F32_16X16X4_F32 (16 tiles of 16x16, K=4).
// Initial W2 global->LDS copy uses the Tensor Data Mover with LDS padding.
// Wave reductions use DPP + v_permlanex16; tanh uses hardware V_TANH_F32.

#define B_ROWS   2048
#define T_STEPS  784
#define H_DIM    64
#define LR       0.01f
#define NSTEP    4
#define W2_LD    66   // even padded stride: 8B-aligned rows, conflict-free banks

typedef float v2f  __attribute__((ext_vector_type(2)));
typedef float v8f  __attribute__((ext_vector_type(8)));
typedef unsigned int v4u __attribute__((ext_vector_type(4)));
typedef int   v4i  __attribute__((ext_vector_type(4)));
typedef int   v8i  __attribute__((ext_vector_type(8)));

#if __has_builtin(__builtin_amdgcn_tanhf)
#define TANH_F32(x) __builtin_amdgcn_tanhf(x)     // hardware V_TANH_F32
#else
#define TANH_F32(x) tanhf(x)
#endif

// ---- wave32 all-lanes sum: DPP xor1/xor2 + row_ror4/8 + permlanex16 ----
__device__ __forceinline__ float wave_sum32(float v) {
    int x;
    x = __builtin_amdgcn_update_dpp(0, __float_as_int(v), 0xB1, 0xF, 0xF, true);  // quad_perm xor1
    v += __int_as_float(x);
    x = __builtin_amdgcn_update_dpp(0, __float_as_int(v), 0x4E, 0xF, 0xF, true);  // quad_perm xor2
    v += __int_as_float(x);
    x = __builtin_amdgcn_update_dpp(0, __float_as_int(v), 0x124, 0xF, 0xF, true); // row_ror:4
    v += __int_as_float(x);
    x = __builtin_amdgcn_update_dpp(0, __float_as_int(v), 0x128, 0xF, 0xF, true); // row_ror:8
    v += __int_as_float(x);
    x = __builtin_amdgcn_permlanex16(__float_as_int(v), __float_as_int(v),
                                     0x76543210, 0xFEDCBA98, false, false);       // cross halves
    v += __int_as_float(x);
    return v;  // full sum in every lane
}

__global__ __launch_bounds__(32)
void ttt_scan_kernel(const float* __restrict__ ts, const float* __restrict__ xs,
                     const unsigned char* __restrict__ mask,
                     const float* __restrict__ W1g, const float* __restrict__ b1g,
                     const float* __restrict__ W2g, const float* __restrict__ b2g,
                     const float* __restrict__ W3g, const float* __restrict__ b3g,
                     float* __restrict__ out)
{
    __shared__ __align__(16) float sW2[H_DIM * W2_LD]; // W2[i][j] at i*66+j
    __shared__ __align__(16) float sH[NSTEP][H_DIM];   // h1_k history (WMMA B staging)
    __shared__ __align__(16) float sG[NSTEP][H_DIM];   // dz2_k history (WMMA A staging)

    const int row = blockIdx.x;
    const int l   = threadIdx.x;           // lane 0..31

    // ---- TDM: async-load this row's initial W2 (64x64 f32) into padded LDS.
    // D# per CDNA5 ISA: 2D tensor, tile 64x64, data_size=4B,
    // pad_enable=1, pad_interval=64 DWORDs (code 5), pad_amount=2 DWORDs (code 1)
    // => LDS rows land at stride 66 floats.
    {
        unsigned long long ga = (unsigned long long)(uintptr_t)W2g;
        unsigned int lds_base = (unsigned int)(uintptr_t)(&sW2[0]); // LDS offset (low 32 bits)
        v4u g0;
        g0[0] = 1u;                                           // count=1 valid descriptor
        g0[1] = lds_base;                                     // lds_addr
        g0[2] = (unsigned int)(ga & 0xFFFFFFFFu);             // global_addr[31:0]
        g0[3] = (unsigned int)((ga >> 32) & 0x01FFFFFFu) | (2u << 30); // addr[56:32] | type=2
        v8i g1;
        g1[0] = (2 << 16) | (1 << 20) | (5 << 22) | (1 << 25); // data_size=4B, pad en, int=64dw, amt=2dw
        g1[1] = (H_DIM & 0xFFFF) << 16;                        // tensor_dim0 = 64
        g1[2] = (H_DIM & 0xFFFF) << 16;                        // tensor_dim1 = 64 (dim0 hi = 0)
        g1[3] = (H_DIM & 0xFFFF) << 16;                        // tile_dim0 = 64 (dim1 hi = 0)
        g1[4] = H_DIM;                                         // tile_dim1 = 64, tile_dim2 = 0
        g1[5] = H_DIM;                                         // tensor_dim0_stride = 64
        g1[6] = (int)((4096u & 0xFFFFu) << 16);                // tensor_dim1_stride lo (unused for 2D)
        g1[7] = 0;
        v4i gz = {0, 0, 0, 0};
        v8i gz8 = {0, 0, 0, 0, 0, 0, 0, 0};
        __builtin_amdgcn_tensor_load_to_lds(g0, g1, gz, gz, gz8, 0);
        __builtin_amdgcn_s_wait_tensorcnt(0);
    }

    // ---- per-row theta kept in VGPRs (lane holds elements l and l+32) ----
    float w1c0lo = W1g[2 * l],         w1c0hi = W1g[2 * (l + 32)];       // W1[:,0]
    float w1c1lo = W1g[2 * l + 1],     w1c1hi = W1g[2 * (l + 32) + 1];   // W1[:,1]
    float b1lo = b1g[l], b1hi = b1g[l + 32];
    float b2lo = b2g[l], b2hi = b2g[l + 32];
    float w3lo = W3g[l], w3hi = W3g[l + 32];
    float b3 = b3g[0];

    const float* xrow = xs + (size_t)row * T_STEPS;
    const unsigned char* mrow = mask + (size_t)row * (T_STEPS - 1);
    float* orow = out + (size_t)row * T_STEPS;

    float x_hat = xrow[0];
    if (l == 0) orow[0] = x_hat;
    __syncthreads();

    for (int t = 1; t < T_STEPS; ++t) {
        const float t_curr = ts[t], t_prev = ts[t - 1];
        const float x_true = xrow[t], x_prev = xrow[t - 1];
        const float x_t = (mrow[t - 1] != 0) ? x_true : x_hat;
        const float u0 = t_curr, u1 = x_prev;
        const float cfac = 1.0f + u0 * u0 + u1 * u1;  // 1 + u.u (z1 dual recurrence)

        float z1lo = fmaf(w1c0lo, u0, fmaf(w1c1lo, u1, b1lo));
        float z1hi = fmaf(w1c0hi, u0, fmaf(w1c1hi, u1, b1hi));

        float hhlo[NSTEP], hhhi[NSTEP], gglo[NSTEP], gghi[NSTEP];
        float sdz1lo = 0.f, sdz1hi = 0.f, sdz2lo = 0.f, sdz2hi = 0.f;

        #pragma unroll
        for (int k = 0; k < NSTEP; ++k) {
            // h1_k = tanh(z1_k); z1 carried in registers via dual recurrence
            float h1lo = TANH_F32(z1lo), h1hi = TANH_F32(z1hi);
            sH[k][l] = h1lo; sH[k][l + 32] = h1hi;
            __syncthreads();

            // v = W2_0 * h1_k : b64 row loads + b64 broadcast, pk-fma friendly
            v2f aLo = {0.f, 0.f}, aHi = {0.f, 0.f};
            #pragma unroll 4
            for (int j = 0; j < H_DIM; j += 2) {
                v2f wl = *(const v2f*)&sW2[l * W2_LD + j];
                v2f wh = *(const v2f*)&sW2[(l + 32) * W2_LD + j];
                v2f hb = *(const v2f*)&sH[k][j];
                aLo += wl * hb;
                aHi += wh * hb;
            }
            float z2lo = aLo.x + aLo.y + b2lo;
            float z2hi = aHi.x + aHi.y + b2hi;
            // exact dual-form corrections: -lr * sum_j (h1_j.h1_k + 1) dz2_j
            #pragma unroll
            for (int j = 0; j < k; ++j) {
                float s = wave_sum32(fmaf(hhlo[j], h1lo, hhhi[j] * h1hi)) + 1.0f;
                float c2 = LR * s;
                z2lo = fmaf(-c2, gglo[j], z2lo);
                z2hi = fmaf(-c2, gghi[j], z2hi);
            }
            float h2lo = TANH_F32(z2lo), h2hi = TANH_F32(z2hi);

            float pred = wave_sum32(fmaf(w3lo, h2lo, w3hi * h2hi)) + b3;
            float gsc = 2.0f * (pred - x_t);          // dL/dpred (mean over 1 elem)
            float dh2lo = gsc * w3lo, dh2hi = gsc * w3hi;   // uses pre-update W3
            float lg = LR * gsc;                      // W3/b3 updated directly
            w3lo = fmaf(-lg, h2lo, w3lo);
            w3hi = fmaf(-lg, h2hi, w3hi);
            b3 -= lg;

            float dz2lo = dh2lo * (1.f - h2lo * h2lo);
            float dz2hi = dh2hi * (1.f - h2hi * h2hi);
            sG[k][l] = dz2lo; sG[k][l + 32] = dz2hi;
            __syncthreads();

            // w = W2_0^T * dz2_k : pair over i, b64 broadcast of dz2, pk-fma
            v2f cLo = {0.f, 0.f}, cHi = {0.f, 0.f};
            #pragma unroll 4
            for (int i = 0; i < H_DIM; i += 2) {
                v2f g2 = *(const v2f*)&sG[k][i];
                v2f wa, wb;
                wa.x = sW2[i * W2_LD + l];        wa.y = sW2[(i + 1) * W2_LD + l];
                wb.x = sW2[i * W2_LD + l + 32];   wb.y = sW2[(i + 1) * W2_LD + l + 32];
                cLo += wa * g2;
                cHi += wb * g2;
            }
            float wlo = cLo.x + cLo.y, whi = cHi.x + cHi.y;
            // corrections: -lr * sum_j (dz2_j.dz2_k) h1_j
            #pragma unroll
            for (int j = 0; j < k; ++j) {
                float s = wave_sum32(fmaf(gglo[j], dz2lo, gghi[j] * dz2hi));
                float c2 = LR * s;
                wlo = fmaf(-c2, hhlo[j], wlo);
                whi = fmaf(-c2, hhhi[j], whi);
            }
            float dz1lo = wlo * (1.f - h1lo * h1lo);
            float dz1hi = whi * (1.f - h1hi * h1hi);

            hhlo[k] = h1lo; hhhi[k] = h1hi; gglo[k] = dz2lo; gghi[k] = dz2hi;
            sdz1lo += dz1lo; sdz1hi += dz1hi;
            sdz2lo += dz2lo; sdz2hi += dz2hi;
            float cf = LR * cfac;
            z1lo = fmaf(-cf, dz1lo, z1lo);   // z1_{k+1} = z1_k - lr(1+u.u)dz1_k
            z1hi = fmaf(-cf, dz1hi, z1hi);
        }

        // ---- end-of-step materialization of W1,b1,b2 (rank-1 in u, u fixed) ----
        w1c0lo = fmaf(-LR * u0, sdz1lo, w1c0lo); w1c0hi = fmaf(-LR * u0, sdz1hi, w1c0hi);
        w1c1lo = fmaf(-LR * u1, sdz1lo, w1c1lo); w1c1hi = fmaf(-LR * u1, sdz1hi, w1c1hi);
        b1lo = fmaf(-LR, sdz1lo, b1lo); b1hi = fmaf(-LR, sdz1hi, b1hi);
        b2lo = fmaf(-LR, sdz2lo, b2lo); b2hi = fmaf(-LR, sdz2hi, b2hi);

        // ---- rank-4 W2 update via WMMA: W2 += (-lr*P) * Q^T, P=[dz2_k], Q=[h1_k]
        // A (16x4 f32): vgpr0: lanes0-15 K=0 / lanes16-31 K=2; vgpr1: K=1 / K=3
        {
            const int half = l >> 4;        // 0: K=0,1   1: K=2,3
            const int lm   = l & 15;
            const int k0   = half ? 2 : 0;
            #pragma unroll
            for (int mb = 0; mb < 4; ++mb) {
                v2f A;
                A.x = -LR * sG[k0][mb * 16 + lm];
                A.y = -LR * sG[k0 + 1][mb * 16 + lm];
                #pragma unroll
                for (int nb = 0; nb < 4; ++nb) {
                    v2f Bm;
                    Bm.x = sH[k0][nb * 16 + lm];
                    Bm.y = sH[k0 + 1][nb * 16 + lm];
                    v8f C;
                    #pragma unroll
                    for (int r = 0; r < 8; ++r)   // C layout: vgpr r -> M=r (+8 hi half)
                        C[r] = sW2[(mb * 16 + half * 8 + r) * W2_LD + nb * 16 + lm];
                    v8f D = __builtin_amdgcn_wmma_f32_16x16x4_f32(
                        false, A, false, Bm, (short)0, C, false, false);
                    #pragma unroll
                    for (int r = 0; r < 8; ++r)
                        sW2[(mb * 16 + half * 8 + r) * W2_LD + nb * 16 + lm] = D[r];
                }
            }
        }
        __syncthreads();

        // ---- final forward with updated theta, input (t_curr + dt, x_true) ----
        const float dt = t_curr - t_prev;
        const float uf0 = t_curr + dt, uf1 = x_true;
        float f1lo = TANH_F32(fmaf(w1c0lo, uf0, fmaf(w1c1lo, uf1, b1lo)));
        float f1hi = TANH_F32(fmaf(w1c0hi, uf0, fmaf(w1c1hi, uf1, b1hi)));
        sH[0][l] = f1lo; sH[0][l + 32] = f1hi;
        __syncthreads();
        v2f aLo = {0.f, 0.f}, aHi = {0.f, 0.f};
        #pragma unroll 4
        for (int j = 0; j < H_DIM; j += 2) {
            v2f wl = *(const v2f*)&sW2[l * W2_LD + j];
            v2f wh = *(const v2f*)&sW2[(l + 32) * W2_LD + j];
            v2f hb = *(const v2f*)&sH[0][j];
            aLo += wl * hb;
            aHi += wh * hb;
        }
        float f2lo = TANH_F32(aLo.x + aLo.y + b2lo), f2hi = TANH_F32(aHi.x + aHi.y + b2hi);
        x_hat = wave_sum32(fmaf(w3lo, f2lo, w3hi * f2hi)) + b3;
        if (l == 0) orow[t] = x_hat;
        __syncthreads();
    }
}

extern "C" void kernel_launch(void* const* d_in, const int* in_sizes, int n_in,
                              void* d_out, int out_size, void* d_ws, size_t ws_size,
                              hipStream_t stream) {
    (void)in_sizes; (void)n_in; (void)out_size; (void)d_ws; (void)ws_size;
    const float*         ts   = (const float*)d_in[0];
    const float*         xs   = (const float*)d_in[1];
    const unsigned char* mask = (const unsigned char*)d_in[2];
    const float*         W1   = (const float*)d_in[3];
    const float*         b1   = (const float*)d_in[4];
    const float*         W2   = (const float*)d_in[5];
    const float*         b2   = (const float*)d_in[6];
    const float*         W3   = (const float*)d_in[7];
    const float*         b3   = (const float*)d_in[8];
    float* out = (float*)d_out;

    hipLaunchKernelGGL(ttt_scan_kernel, dim3(B_ROWS), dim3(32), 0, stream,
                       ts, xs, mask, W1, b1, W2, b2, W3, b3, out);
}